// GCN_77120432767264
// MI455X (gfx1250) — compile-verified
//
#include <hip/hip_runtime.h>
#include <hip/hip_bf16.h>

typedef __attribute__((ext_vector_type(2))) float f32x2;
typedef __attribute__((ext_vector_type(8))) float f32x8;

#define HDIM 64

__device__ __forceinline__ void atomic_add_f32(float* p, float v) {
  unsafeAtomicAdd(p, v);   // hardware global_atomic_add_f32
}

// ---------------- degree / normalization ----------------
__global__ void k_deg_init(float* deg, int N) {
  int i = blockIdx.x * blockDim.x + threadIdx.x;
  if (i < N) deg[i] = 1.0f;  // self-loop contribution
}
__global__ void k_deg_count(const int* __restrict__ col, float* deg, int E) {
  int e = blockIdx.x * blockDim.x + threadIdx.x;
  if (e < E) atomic_add_f32(&deg[col[e]], 1.0f);
}
__global__ void k_dinv(float* deg, int N) {
  int i = blockIdx.x * blockDim.x + threadIdx.x;
  if (i < N) {
    float d = deg[i];
    deg[i] = (d > 0.0f) ? rsqrtf(d) : 0.0f;
  }
}

// ---------------- fp32 WMMA GEMM: XW = X(N x K) @ W(K x 64) ----------------
// grid.x = N/16 row tiles; 128 threads = 4 waves, wave w handles col tile [16w,16w+16)
// A frag (16x4 f32): lane l -> row m0+(l&15), K pair ((l>>4)*2, +1)
// B frag (4x16 f32): lane l -> col n0+(l&15), same K pair
// C/D (16x16 f32, 8 VGPRs): elem[v] = D[8*(l>>4)+v][l&15]
__global__ void k_gemm_wmma(const float* __restrict__ X, const float* __restrict__ W,
                            float* __restrict__ XW, int K) {
  const int lane  = threadIdx.x & 31;
  const int wave  = threadIdx.x >> 5;
  const int m0    = blockIdx.x << 4;
  const int n0    = wave << 4;
  const int l16   = lane & 15;
  const int hi    = lane >> 4;     // 0 or 1
  const int khalf = hi << 1;       // 0 or 2
  const float* __restrict__ arow = X + (size_t)(m0 + l16) * K;
  f32x8 acc = {};
  for (int k = 0; k < K; k += 4) {
    f32x2 a, b;
    a.x = arow[k + khalf];
    a.y = arow[k + khalf + 1];
    b.x = W[(size_t)(k + khalf) * HDIM + n0 + l16];
    b.y = W[(size_t)(k + khalf + 1) * HDIM + n0 + l16];
    acc = __builtin_amdgcn_wmma_f32_16x16x4_f32(false, a, false, b, (short)0, acc,
                                                false, false);
  }
  const int mbase = m0 + (hi << 3);
  float* __restrict__ dst = XW + (size_t)mbase * HDIM + n0 + l16;
#pragma unroll
  for (int v = 0; v < 8; ++v) dst[(size_t)v * HDIM] = acc[v];
}

// scalar tail for N % 16 != 0 (not used for N=100000, kept for robustness)
__global__ void k_gemm_tail(const float* __restrict__ X, const float* __restrict__ W,
                            float* __restrict__ XW, int K, int mStart, int mEnd) {
  int idx = blockIdx.x * blockDim.x + threadIdx.x;
  int rows = mEnd - mStart;
  if (idx >= rows * HDIM) return;
  int m = mStart + idx / HDIM, n = idx % HDIM;
  float acc = 0.0f;
  for (int k = 0; k < K; ++k) acc += X[(size_t)m * K + k] * W[(size_t)k * HDIM + n];
  XW[(size_t)m * HDIM + n] = acc;
}

// ---------------- bias broadcast init: out[n][h] = b[h] ----------------
__global__ void k_bias_init(float* __restrict__ out, const float* __restrict__ b,
                            size_t total) {
  size_t i = (size_t)blockIdx.x * blockDim.x + threadIdx.x;
  if (i < total) out[i] = b[i & (HDIM - 1)];
}

// ---------------- edge scatter: out[c] += xw[r] * dinv[r]*dinv[c] ----------------
// one wave32 per edge (edges E..E+N-1 are the self-loops); lane handles 2 features
__global__ void k_scatter(const float* __restrict__ xw, const int* __restrict__ row,
                          const int* __restrict__ col, const float* __restrict__ dinv,
                          float* __restrict__ out, int E, int N) {
  int e    = blockIdx.x * (blockDim.x >> 5) + (threadIdx.x >> 5);
  int lane = threadIdx.x & 31;
  if (e >= E + N) return;
  int r, c;
  if (e < E) { r = row[e]; c = col[e]; }
  else       { r = e - E;  c = r; }
  float w = dinv[r] * dinv[c];
  f32x2 v = *(const f32x2*)(xw + (size_t)r * HDIM + 2 * lane);
  float* dst = out + (size_t)c * HDIM + 2 * lane;
  atomic_add_f32(dst,     v.x * w);
  atomic_add_f32(dst + 1, v.y * w);
}

// ---------------- global mean pool ----------------
__global__ void k_pool_init(float* sums, float* cnts, int G) {
  int i = blockIdx.x * blockDim.x + threadIdx.x;
  if (i < G * HDIM) sums[i] = 0.0f;
  if (i < G)        cnts[i] = 0.0f;
}
__global__ void k_pool(const float* __restrict__ h, const int* __restrict__ batch,
                       float* __restrict__ sums, float* __restrict__ cnts, int N) {
  int n    = blockIdx.x * (blockDim.x >> 5) + (threadIdx.x >> 5);
  int lane = threadIdx.x & 31;
  if (n >= N) return;
  int g = batch[n];
  f32x2 v = *(const f32x2*)(h + (size_t)n * HDIM + 2 * lane);
  float* dst = sums + (size_t)g * HDIM + 2 * lane;
  atomic_add_f32(dst,     v.x);
  atomic_add_f32(dst + 1, v.y);
  if (lane == 0) atomic_add_f32(&cnts[g], 1.0f);
}

// ---------------- final linear: out = (sums/cnt) @ Wl + bl ----------------
__global__ void k_final(const float* __restrict__ sums, const float* __restrict__ cnts,
                        const float* __restrict__ Wl, const float* __restrict__ bl,
                        float* __restrict__ out, int G, int C) {
  int i = blockIdx.x * blockDim.x + threadIdx.x;
  if (i >= G * C) return;
  int g = i / C, c = i % C;
  float inv = 1.0f / fmaxf(cnts[g], 1.0f);
  float acc = bl[c];
  const float* s = sums + (size_t)g * HDIM;
#pragma unroll 8
  for (int k = 0; k < HDIM; ++k) acc += s[k] * inv * Wl[k * C + c];
  out[i] = acc;
}

extern "C" void kernel_launch(void* const* d_in, const int* in_sizes, int n_in,
                              void* d_out, int out_size, void* d_ws, size_t ws_size,
                              hipStream_t stream) {
  const float* x     = (const float*)d_in[0];
  const int*   ei    = (const int*)d_in[1];
  const int*   batch = (const int*)d_in[2];
  const float* W1 = (const float*)d_in[3];
  const float* b1 = (const float*)d_in[4];
  const float* W2 = (const float*)d_in[5];
  const float* b2 = (const float*)d_in[6];
  const float* W3 = (const float*)d_in[7];
  const float* b3 = (const float*)d_in[8];
  const float* Wl = (const float*)d_in[9];
  const float* bl = (const float*)d_in[10];

  const int N = in_sizes[2];
  const int E = in_sizes[1] / 2;
  const int D = in_sizes[0] / N;   // 128
  const int C = in_sizes[10];      // 10
  const int G = out_size / C;      // 1024

  const int* row = ei;
  const int* col = ei + E;

  // workspace carve-out: dinv | buf0 (xw) | buf1 (h) | pooled sums | counts
  char* ws = (char*)d_ws;
  auto align256 = [](size_t o) { return (o + 255) & ~(size_t)255; };
  size_t off = 0;
  float* dinv = (float*)(ws + off); off = align256(off + (size_t)N * 4);
  float* buf0 = (float*)(ws + off); off = align256(off + (size_t)N * HDIM * 4);
  float* buf1 = (float*)(ws + off); off = align256(off + (size_t)N * HDIM * 4);
  float* sums = (float*)(ws + off); off = align256(off + (size_t)G * HDIM * 4);
  float* cnts = (float*)(ws + off); off = align256(off + (size_t)G * 4);
  (void)ws_size; (void)n_in;

  const int TB = 256;
  const int wavesPerBlk = TB / 32;

  // normalization coefficients (shared by all layers)
  k_deg_init <<<(N + TB - 1) / TB, TB, 0, stream>>>(dinv, N);
  k_deg_count<<<(E + TB - 1) / TB, TB, 0, stream>>>(col, dinv, E);
  k_dinv     <<<(N + TB - 1) / TB, TB, 0, stream>>>(dinv, N);

  const int rowTiles   = N / 16;                       // 6250 for N=100000
  const int tailRows   = N - rowTiles * 16;            // 0 for N=100000
  const int edgeBlocks = (E + N + wavesPerBlk - 1) / wavesPerBlk;
  const size_t nh      = (size_t)N * HDIM;
  const int nhBlocks   = (int)((nh + TB - 1) / TB);

  const float* Ws[3] = {W1, W2, W3};
  const float* bs[3] = {b1, b2, b3};
  for (int layer = 0; layer < 3; ++layer) {
    const float* src = (layer == 0) ? x : buf1;
    const int    K   = (layer == 0) ? D : HDIM;
    if (rowTiles > 0)
      k_gemm_wmma<<<rowTiles, 128, 0, stream>>>(src, Ws[layer], buf0, K);
    if (tailRows > 0)
      k_gemm_tail<<<(tailRows * HDIM + TB - 1) / TB, TB, 0, stream>>>(
          src, Ws[layer], buf0, K, rowTiles * 16, N);
    k_bias_init<<<nhBlocks, TB, 0, stream>>>(buf1, bs[layer], nh);
    k_scatter  <<<edgeBlocks, TB, 0, stream>>>(buf0, row, col, dinv, buf1, E, N);
  }

  // global mean pool + classifier head
  k_pool_init<<<(G * HDIM + TB - 1) / TB, TB, 0, stream>>>(sums, cnts, G);
  k_pool     <<<(N + wavesPerBlk - 1) / wavesPerBlk, TB, 0, stream>>>(buf1, batch, sums, cnts, N);
  k_final    <<<(G * C + TB - 1) / TB, TB, 0, stream>>>(sums, cnts, Wl, bl, (float*)d_out, G, C);
}